// CrowdCountingLoss_46162308497941
// MI455X (gfx1250) — compile-verified
//
#include <hip/hip_runtime.h>
#include <math.h>

// CDNA5 / gfx1250 implementation of CrowdCountingLoss (MSE + count MSE + unbalanced
// debiased Sinkhorn divergence, geomloss-style eps-scaling).
//
// Heavy compute: batched 768^3 f32 GEMMs via V_WMMA_F32_16X16X4_F32 (uniform scalar
// k-loop -> EXEC all-ones, 4 b64 loads per 4 WMMAs); the ~50 eps-scaling softmin
// sweeps stream the cost matrices out of the 192MB L2. If workspace permits
// (~76MB) we also materialize C_yx so every softmin sweep is a coalesced row
// stream (no 16x strided-read L2 penalty); else fall back to transposed reads.
// Data-dependent eps schedule is built on-device (diameter via deterministic
// integer-bit atomicMin/Max on non-negative data); fixed MAX_EPS launches with
// device-side guards keep the launch graph static and deterministic.

typedef float v2f __attribute__((ext_vector_type(2)));
typedef float v8f __attribute__((ext_vector_type(8)));

#define NPTS 768
#define BATCH 8
#define SMAT (768 * 768)
#define BSMAT ((size_t)BATCH * SMAT)
#define NROWS (BATCH * NPTS)          // 6144
#define MAX_EPS 64
#define RHO 0.01f                     // reach^2
#define BLUR2 0.04f                   // blur^2
#define BASELOG (-6.6438561897747395f) // -ln(768)

// ---------------- per-row stats: ||row||^2, row sums (counts), row SSE ----------
__global__ void k_rowstats(const float* __restrict__ pred, const float* __restrict__ gt,
                           float* __restrict__ x2, float* __restrict__ y2,
                           float* __restrict__ rsx, float* __restrict__ rsy,
                           float* __restrict__ rsse) {
  int row = blockIdx.x;
  int t = threadIdx.x;
  const float* xr = pred + (size_t)row * NPTS;
  const float* yr = gt + (size_t)row * NPTS;
  float sx = 0.f, sy = 0.f, sxx = 0.f, syy = 0.f, se = 0.f;
  for (int j = t; j < NPTS; j += 256) {
    float xv = xr[j], yv = yr[j];
    sx += xv; sy += yv; sxx += xv * xv; syy += yv * yv;
    float dd = xv - yv; se += dd * dd;
  }
  __shared__ float red[256];
  auto reduce = [&](float v) -> float {
    red[t] = v; __syncthreads();
    for (int s = 128; s > 0; s >>= 1) { if (t < s) red[t] += red[t + s]; __syncthreads(); }
    float r = red[0]; __syncthreads(); return r;
  };
  float tsx = reduce(sx), tsy = reduce(sy), tsxx = reduce(sxx), tsyy = reduce(syy), tse = reduce(se);
  if (t == 0) { x2[row] = tsxx; y2[row] = tsyy; rsx[row] = tsx; rsy[row] = tsy; rsse[row] = tse; }
}

// ---------------- per-dim min/max for diameter (data is non-negative) -----------
__global__ void k_init_minmax(int* __restrict__ colmin, int* __restrict__ colmax) {
  int d = blockIdx.x * 256 + threadIdx.x;
  if (d < NPTS) { colmin[d] = 0x7F800000; colmax[d] = 0; }
}

__global__ void k_minmax(const float* __restrict__ pred, const float* __restrict__ gt,
                         int* __restrict__ colmin, int* __restrict__ colmax) {
  int r0 = blockIdx.x * 32;           // 384 blocks cover 12288 rows (pred then gt)
  int t = threadIdx.x;
  float mn[3] = {INFINITY, INFINITY, INFINITY};
  float mx[3] = {0.f, 0.f, 0.f};
  for (int r = r0; r < r0 + 32; ++r) {
    const float* src = (r < NROWS) ? (pred + (size_t)r * NPTS)
                                   : (gt + (size_t)(r - NROWS) * NPTS);
#pragma unroll
    for (int c = 0; c < 3; c++) {
      float v = src[t + c * 256];
      mn[c] = fminf(mn[c], v); mx[c] = fmaxf(mx[c], v);
    }
  }
#pragma unroll
  for (int c = 0; c < 3; c++) {       // int-bit compare valid for non-negative floats
    int d = t + c * 256;
    atomicMin(&colmin[d], __float_as_int(mn[c]));
    atomicMax(&colmax[d], __float_as_int(mx[c]));
  }
}

// ---------------- scalar setup: diameter, eps schedule, density+count loss ------
__global__ void k_setup(const int* __restrict__ colmin, const int* __restrict__ colmax,
                        const float* __restrict__ rsx, const float* __restrict__ rsy,
                        const float* __restrict__ rsse,
                        float* __restrict__ eps_list, int* __restrict__ n_eps,
                        float* __restrict__ loss_base) {
  int t = threadIdx.x;
  __shared__ float red[256];
  __shared__ float pc_s[BATCH], gc_s[BATCH];
  auto reduce = [&](float v) -> float {
    red[t] = v; __syncthreads();
    for (int s = 128; s > 0; s >>= 1) { if (t < s) red[t] += red[t + s]; __syncthreads(); }
    float r = red[0]; __syncthreads(); return r;
  };
  float acc = 0.f;
#pragma unroll
  for (int c = 0; c < 3; c++) {
    int d = t + c * 256;
    float r = __int_as_float(colmax[d]) - __int_as_float(colmin[d]);
    acc += r * r;
  }
  float diam2 = reduce(acc);
  for (int b = 0; b < BATCH; b++) {
    float s = 0.f; for (int i = t; i < NPTS; i += 256) s += rsx[b * NPTS + i];
    float tot = reduce(s); if (t == 0) pc_s[b] = tot;
    s = 0.f; for (int i = t; i < NPTS; i += 256) s += rsy[b * NPTS + i];
    tot = reduce(s); if (t == 0) gc_s[b] = tot;
  }
  float se = 0.f;
  for (int i = t; i < NROWS; i += 256) se += rsse[i];
  float sse_tot = reduce(se);
  if (t == 0) {
    float closs = 0.f;
    for (int b = 0; b < BATCH; b++) { float d = pc_s[b] - gc_s[b]; closs += d * d; }
    closs *= (1.0f / BATCH);
    float dens = sse_tot / (float)((size_t)BATCH * SMAT);
    *loss_base = dens + closs;
    float diam = sqrtf(diam2);
    eps_list[0] = diam2;                         // diameter^p
    int n = 1;
    float v = 2.0f * logf(diam);                 // arange(2 ln d, 2 ln blur, 2 ln scaling)
    float stop = 2.0f * logf(0.2f);
    float step = 2.0f * logf(0.9f);
    while (v > stop && n < MAX_EPS - 1) { eps_list[n++] = expf(v); v += step; }
    eps_list[n++] = BLUR2;
    *n_eps = n;
  }
}

// ---------------- WMMA f32 GEMM + cost epilogue ---------------------------------
// C = max(||a_i||^2 + ||b_j||^2 - 2 a.b^T, 0) / 2 ; 128x64 block tile, 32x32/wave.
// blockIdx.z encodes (matrix, batch): mat 0:xy 1:xx 2:yy 3:yx(optional).
__global__ void __launch_bounds__(256) k_gemm(const float* __restrict__ pred,
                                              const float* __restrict__ gt,
                                              const float* __restrict__ x2,
                                              const float* __restrict__ y2,
                                              float* __restrict__ Cbase) {
  int z = blockIdx.z;
  int mat = z >> 3;                // 0:xy 1:xx 2:yy 3:yx
  int b = z & 7;
  const float* A; const float* Bm; const float* rn; const float* cn;
  if (mat == 0)      { A = pred; Bm = gt;   rn = x2; cn = y2; }
  else if (mat == 1) { A = pred; Bm = pred; rn = x2; cn = x2; }
  else if (mat == 2) { A = gt;   Bm = gt;   rn = y2; cn = y2; }
  else               { A = gt;   Bm = pred; rn = y2; cn = x2; }
  float* Cout = Cbase + (size_t)mat * BSMAT + (size_t)b * SMAT;
  A += (size_t)b * SMAT; Bm += (size_t)b * SMAT;
  rn += b * NPTS; cn += b * NPTS;

  int wave = threadIdx.x >> 5;
  int lane = threadIdx.x & 31;
  int lr = lane & 15;
  int kh = (lane >> 4) << 1;           // per-ISA A/B fragment: lanes 16-31 hold K+2
  int wr = wave & 3, wc = wave >> 2;
  int m0 = blockIdx.y * 128 + wr * 32;
  int n0 = blockIdx.x * 64 + wc * 32;
  // fold lane-dependent k-offset into the bases -> uniform scalar k-loop below
  const float* Ar0 = A + (size_t)(m0 + lr) * NPTS + kh;
  const float* Ar1 = Ar0 + 16 * NPTS;
  const float* Br0 = Bm + (size_t)(n0 + lr) * NPTS + kh;
  const float* Br1 = Br0 + 16 * NPTS;

  v8f acc00 = {}, acc01 = {}, acc10 = {}, acc11 = {};
  for (int kk = 0; kk < NPTS; kk += 4) {   // uniform trip count: EXEC stays all-ones
    v2f a0 = *(const v2f*)(Ar0 + kk);
    v2f a1 = *(const v2f*)(Ar1 + kk);
    v2f b0 = *(const v2f*)(Br0 + kk);
    v2f b1 = *(const v2f*)(Br1 + kk);
    acc00 = __builtin_amdgcn_wmma_f32_16x16x4_f32(false, a0, false, b0, (short)0, acc00, false, false);
    acc01 = __builtin_amdgcn_wmma_f32_16x16x4_f32(false, a0, false, b1, (short)0, acc01, false, false);
    acc10 = __builtin_amdgcn_wmma_f32_16x16x4_f32(false, a1, false, b0, (short)0, acc10, false, false);
    acc11 = __builtin_amdgcn_wmma_f32_16x16x4_f32(false, a1, false, b1, (short)0, acc11, false, false);
  }

  int colofs = lane & 15;
  int rofs = (lane >> 4) << 3;         // C/D layout: lanes 16-31 hold rows M+8
  auto store_tile = [&](v8f accv, int bm, int bn) {
    int col = bn + colofs;
    float cv = cn[col];
#pragma unroll
    for (int r = 0; r < 8; r++) {
      int row = bm + rofs + r;
      float val = rn[row] + cv - 2.0f * accv[r];
      Cout[(size_t)row * NPTS + col] = fmaxf(val, 0.0f) * 0.5f;
    }
  };
  store_tile(acc00, m0, n0);
  store_tile(acc01, m0, n0 + 16);
  store_tile(acc10, m0 + 16, n0);
  store_tile(acc11, m0 + 16, n0 + 16);
}

// ---------------- softmin: f_i = -eps * LSE_j( base + g_j/eps - C_ij/eps ) ------
// mode 0: out = d*sm (init, g=null)   mode 1: out = 0.5*(f_old + d*sm), guarded
// mode 2: out = d*sm at eps=blur^2 (final extrapolation)
__global__ void k_softmin(const float* __restrict__ C, const float* __restrict__ g_in,
                          const float* __restrict__ f_old, float* __restrict__ f_out,
                          const float* __restrict__ eps_list, const int* __restrict__ n_eps_ptr,
                          int iter, int mode, int transpose) {
  int rowg = blockIdx.x;               // 0..6143
  int b = rowg / NPTS;
  int i = rowg - b * NPTS;
  int t = threadIdx.x;
  int ne = *n_eps_ptr;
  if (mode == 1 && iter >= ne) {       // inactive over-count iteration: pass through
    if (t == 0) f_out[rowg] = f_old[rowg];
    return;
  }
  float eps = (mode == 2) ? BLUR2 : eps_list[iter < ne ? iter : ne - 1];
  float inv_eps = 1.0f / eps;
  float damp = 1.0f / (1.0f + eps / RHO);
  const float* Cb = C + (size_t)b * SMAT;
  const float* gb = g_in ? (g_in + b * NPTS) : nullptr;

  float v[3];
#pragma unroll
  for (int c = 0; c < 3; c++) {
    int j = t + c * 256;
    float hj = BASELOG + (gb ? gb[j] * inv_eps : 0.0f);
    float cv = transpose ? Cb[(size_t)j * NPTS + i] : Cb[(size_t)i * NPTS + j];
    v[c] = hj - cv * inv_eps;
  }
  __shared__ float red[256];
  float lm = fmaxf(fmaxf(v[0], v[1]), v[2]);
  red[t] = lm; __syncthreads();
  for (int s = 128; s > 0; s >>= 1) { if (t < s) red[t] = fmaxf(red[t], red[t + s]); __syncthreads(); }
  float m = red[0]; __syncthreads();
  float ls = expf(v[0] - m) + expf(v[1] - m) + expf(v[2] - m);
  red[t] = ls; __syncthreads();
  for (int s = 128; s > 0; s >>= 1) { if (t < s) red[t] += red[t + s]; __syncthreads(); }
  if (t == 0) {
    float sm = -eps * (m + logf(red[0]));
    f_out[rowg] = (mode == 1) ? 0.5f * (f_old[rowg] + damp * sm) : damp * sm;
  }
}

// ---------------- final divergence + total loss ---------------------------------
__global__ void k_final(const float* __restrict__ faa, const float* __restrict__ fba,
                        const float* __restrict__ gbb, const float* __restrict__ gab,
                        const float* __restrict__ loss_base, float* __restrict__ out) {
  int t = threadIdx.x;
  float acc = 0.f;
  for (int i = t; i < NROWS; i += 256) {
    acc += expf(-faa[i] * 100.0f) - expf(-fba[i] * 100.0f);   // 1/rho = 100
    acc += expf(-gbb[i] * 100.0f) - expf(-gab[i] * 100.0f);
  }
  __shared__ float red[256];
  red[t] = acc; __syncthreads();
  for (int s = 128; s > 0; s >>= 1) { if (t < s) red[t] += red[t + s]; __syncthreads(); }
  if (t == 0) {
    float w = RHO + BLUR2 * 0.5f;                 // 0.03
    float spatial = w * red[0] / (float)(NPTS * BATCH); // weights 1/N, mean over B
    out[0] = loss_base[0] + spatial;              // ALPHA = 1
  }
}

// ---------------- host orchestration --------------------------------------------
extern "C" void kernel_launch(void* const* d_in, const int* in_sizes, int n_in,
                              void* d_out, int out_size, void* d_ws, size_t ws_size,
                              hipStream_t stream) {
  (void)in_sizes; (void)n_in; (void)out_size;
  const float* pred = (const float*)d_in[0];
  const float* gt = (const float*)d_in[1];  // d_in[2] (gt_blur_map) unused by the loss

  // workspace layout (floats): small buffers first, matrices after (so the layout
  // is identical whether or not the optional 4th matrix C_yx fits).
  float* ws = (float*)d_ws;
  float* x2   = ws;
  float* y2   = x2 + NROWS;
  float* rsx  = y2 + NROWS;
  float* rsy  = rsx + NROWS;
  float* rsse = rsy + NROWS;
  float* fbuf = rsse + NROWS;                      // 12 * 6144 potentials
  float* f_ba[2] = {fbuf,             fbuf + NROWS};
  float* g_ab[2] = {fbuf + 2 * NROWS, fbuf + 3 * NROWS};
  float* f_aa[2] = {fbuf + 4 * NROWS, fbuf + 5 * NROWS};
  float* g_bb[2] = {fbuf + 6 * NROWS, fbuf + 7 * NROWS};
  float* f_ba_f = fbuf + 8 * NROWS;
  float* g_ab_f = fbuf + 9 * NROWS;
  float* f_aa_f = fbuf + 10 * NROWS;
  float* g_bb_f = fbuf + 11 * NROWS;
  int* colmin = (int*)(fbuf + 12 * NROWS);
  int* colmax = colmin + NPTS;
  float* eps_list = (float*)(colmax + NPTS);
  int* n_eps = (int*)(eps_list + MAX_EPS);
  float* loss_base = (float*)(n_eps + 4);
  float* Cbase = (float*)(loss_base + 4);          // 3 or 4 cost matrices

  size_t small_floats = (size_t)(Cbase - ws);
  bool use_yx = ws_size >= (small_floats + 4 * BSMAT) * sizeof(float);

  float* Cxy = Cbase;
  float* Cxx = Cbase + BSMAT;
  float* Cyy = Cbase + 2 * BSMAT;
  float* Cyx = Cbase + 3 * BSMAT;                  // only valid if use_yx
  // yx direction: either read materialized C_yx row-wise, or C_xy transposed
  const float* Cyx_src = use_yx ? Cyx : Cxy;
  int yx_tr = use_yx ? 0 : 1;

  k_init_minmax<<<3, 256, 0, stream>>>(colmin, colmax);
  k_rowstats<<<NROWS, 256, 0, stream>>>(pred, gt, x2, y2, rsx, rsy, rsse);
  k_minmax<<<384, 256, 0, stream>>>(pred, gt, colmin, colmax);
  k_setup<<<1, 256, 0, stream>>>(colmin, colmax, rsx, rsy, rsse, eps_list, n_eps, loss_base);

  dim3 ggrid(NPTS / 64, NPTS / 128, (use_yx ? 4 : 3) * BATCH);
  k_gemm<<<ggrid, 256, 0, stream>>>(pred, gt, x2, y2, Cbase);

  // initial potentials at eps_list[0] (h = uniform log weights)
  k_softmin<<<NROWS, 256, 0, stream>>>(Cxx, nullptr, nullptr, f_aa[0], eps_list, n_eps, 0, 0, 0);
  k_softmin<<<NROWS, 256, 0, stream>>>(Cyy, nullptr, nullptr, g_bb[0], eps_list, n_eps, 0, 0, 0);
  k_softmin<<<NROWS, 256, 0, stream>>>(Cyx_src, nullptr, nullptr, g_ab[0], eps_list, n_eps, 0, 0, yx_tr);
  k_softmin<<<NROWS, 256, 0, stream>>>(Cxy, nullptr, nullptr, f_ba[0], eps_list, n_eps, 0, 0, 0);

  // eps-scaling loop, double-buffered; inactive iterations copy (guard in-kernel)
  for (int it = 0; it < MAX_EPS; ++it) {
    int p = it & 1, q = p ^ 1;
    k_softmin<<<NROWS, 256, 0, stream>>>(Cxy, g_ab[p], f_ba[p], f_ba[q], eps_list, n_eps, it, 1, 0);
    k_softmin<<<NROWS, 256, 0, stream>>>(Cyx_src, f_ba[p], g_ab[p], g_ab[q], eps_list, n_eps, it, 1, yx_tr);
    k_softmin<<<NROWS, 256, 0, stream>>>(Cxx, f_aa[p], f_aa[p], f_aa[q], eps_list, n_eps, it, 1, 0);
    k_softmin<<<NROWS, 256, 0, stream>>>(Cyy, g_bb[p], g_bb[p], g_bb[q], eps_list, n_eps, it, 1, 0);
  }

  // final extrapolation at eps = blur^2 (buffers at parity 0: MAX_EPS is even)
  k_softmin<<<NROWS, 256, 0, stream>>>(Cxy, g_ab[0], nullptr, f_ba_f, eps_list, n_eps, MAX_EPS, 2, 0);
  k_softmin<<<NROWS, 256, 0, stream>>>(Cyx_src, f_ba[0], nullptr, g_ab_f, eps_list, n_eps, MAX_EPS, 2, yx_tr);
  k_softmin<<<NROWS, 256, 0, stream>>>(Cxx, f_aa[0], nullptr, f_aa_f, eps_list, n_eps, MAX_EPS, 2, 0);
  k_softmin<<<NROWS, 256, 0, stream>>>(Cyy, g_bb[0], nullptr, g_bb_f, eps_list, n_eps, MAX_EPS, 2, 0);

  k_final<<<1, 256, 0, stream>>>(f_aa_f, f_ba_f, g_bb_f, g_ab_f, loss_base, (float*)d_out);
}